// PostProcessor2D_75651553952298
// MI455X (gfx1250) — compile-verified
//
#include <hip/hip_runtime.h>
#include <hip/hip_bf16.h>

#define HH 4096
#define WW 4096
#define KTOP 4096
#define NBINS 16384
#define SELCAP 8192

// ---------------- workspace layout (32-bit words on d_ws) ----------------
// W[0]            : candidate counter (all peaks)
// W[1]            : selected counter (peaks with bin >= cutoff)
// W[2]            : cutoff bin
// W[3]            : pad
// W[4 .. 16388)   : histogram (16384 bins over (0.9, 1.0])
// W[16388..24580) : selected values  (8192 f32)
// W[24580..32772) : selected indices (8192 u32)
// W[32772.. )     : candidate values (cap f32) then candidate indices (cap u32)
#define WS_HIST 4
#define WS_SELV 16388
#define WS_SELI 24580
#define WS_CAND 32772

typedef int v4i __attribute__((vector_size(16)));

// ---------------- CDNA5 async global->LDS helpers ----------------
__device__ __forceinline__ void async_copy_b128(const void* gsrc, void* ldst) {
#if __has_builtin(__builtin_amdgcn_global_load_async_to_lds_b128)
    __builtin_amdgcn_global_load_async_to_lds_b128(
        (__attribute__((address_space(1))) v4i*)gsrc,
        (__attribute__((address_space(3))) v4i*)ldst,
        0, 0);
#else
    unsigned l = (unsigned)(unsigned long long)(__attribute__((address_space(3))) char*)ldst;
    unsigned long long g = (unsigned long long)gsrc;
    asm volatile("global_load_async_to_lds_b128 %0, %1, off"
                 :: "v"(l), "v"(g) : "memory");
#endif
}

__device__ __forceinline__ void wait_async0() {
#if __has_builtin(__builtin_amdgcn_s_wait_asynccnt)
    __builtin_amdgcn_s_wait_asynccnt(0);
#else
    asm volatile("s_wait_asynccnt 0" ::: "memory");
#endif
}

// ---------------- kernel 0: zero counters + histogram ----------------
__global__ void __launch_bounds__(1024) init_kernel(unsigned* __restrict__ Wp) {
    unsigned i = blockIdx.x * blockDim.x + threadIdx.x;
    if (i < WS_HIST + NBINS) Wp[i] = 0u;
}

// ---------------- kernel 1: threshold + 7x7 NMS maxpool + candidate scan ----
// Tile: 128 cols x 32 rows per block, halo 3. Raw tile streamed via async
// global->LDS b128 copies (cols [x0-4, x0+132) = 34 aligned b128 chunks/row,
// rows [y0-3, y0+35) = 38 rows). Separable max: row pass -> hx, col pass.
__global__ void __launch_bounds__(256) conf_kernel(const float* __restrict__ pred,
                                                   float* __restrict__ conf,
                                                   unsigned* __restrict__ Wp,
                                                   unsigned cap) {
    const int TW = 128, TH = 32, LROWS = TH + 6, LW = 136, NCH = 34;
    __shared__ float raw[LROWS][LW];   // 38*136*4 = 20672 B
    __shared__ float hx[LROWS][TW];    // 38*128*4 = 19456 B

    const int x0 = blockIdx.x * TW;
    const int y0 = blockIdx.y * TH;
    const int tid = threadIdx.x;

    // ---- stage 1: async-stream raw tile (with halo) into LDS ----
    const int TOT = LROWS * NCH; // 1292 b128 chunks
    for (int id = tid; id < TOT; id += 256) {
        int r = id / NCH, c = id - r * NCH;
        int gy = y0 - 3 + r;  gy = gy < 0 ? 0 : (gy > HH - 1 ? HH - 1 : gy);
        int gx = x0 - 4 + c * 4; gx = gx < 0 ? 0 : (gx > WW - 4 ? WW - 4 : gx);
        async_copy_b128(pred + (size_t)gy * WW + gx, &raw[r][c * 4]);
    }
    wait_async0();
    __syncthreads();

    // ---- stage 2: horizontal 7-tap max (threshold applied on read) ----
    const int HT = LROWS * TW; // 4864
    for (int e = tid; e < HT; e += 256) {
        int r = e >> 7;
        int c = e & 127;
        int gy = y0 - 3 + r;
        float m = 0.0f;
        if (gy >= 0 && gy < HH) {
#pragma unroll
            for (int dx = 0; dx < 7; ++dx) {
                int gx = x0 + c - 3 + dx;
                float v = raw[r][c + 1 + dx];
                v = (gx >= 0 && gx < WW && v > 0.9f) ? v : 0.0f;
                m = fmaxf(m, v);
            }
        }
        hx[r][c] = m;
    }
    __syncthreads();

    // ---- stage 3: vertical 7-tap max, peak test, store + candidate push ----
    const int cg = tid & 31;   // 32 groups of 4 cols
    const int r0 = tid >> 5;   // 8 row phases
#pragma unroll
    for (int k = 0; k < 4; ++k) {
        int ro = r0 + k * 8;          // 0..31
        int gy = y0 + ro;
        float ov[4];
#pragma unroll
        for (int q = 0; q < 4; ++q) {
            int c = cg * 4 + q;
            float m = 0.0f;
#pragma unroll
            for (int dy = 0; dy < 7; ++dy) m = fmaxf(m, hx[ro + dy][c]);
            float ctr = raw[ro + 3][c + 4];
            float ct = (ctr > 0.9f) ? ctr : 0.0f;
            float o = (ct > 0.0f && m <= ct) ? ct : 0.0f;
            ov[q] = o;
            if (o > 0.0f) {
                int bin = (int)((o - 0.9f) * 163840.0f); // 16384 / 0.1
                bin = bin < 0 ? 0 : (bin > NBINS - 1 ? NBINS - 1 : bin);
                atomicAdd(Wp + WS_HIST + bin, 1u);
                unsigned p = atomicAdd(Wp, 1u);
                if (p < cap) {
                    float* candV = (float*)(Wp + WS_CAND);
                    unsigned* candI = Wp + WS_CAND + cap;
                    candV[p] = o;
                    candI[p] = (unsigned)(gy * WW + x0 + c);
                }
            }
        }
        float4* dst = (float4*)(conf + (size_t)gy * WW + x0 + cg * 4);
        *dst = make_float4(ov[0], ov[1], ov[2], ov[3]);
    }
}

// ---------------- kernel 2: suffix-scan histogram -> k-th value cutoff bin --
__global__ void __launch_bounds__(512) cutoff_kernel(unsigned* __restrict__ Wp) {
    __shared__ unsigned csum[512];
    const unsigned t = threadIdx.x;
    const unsigned* hist = Wp + WS_HIST;

    unsigned s = 0;
    for (int u = 0; u < 32; ++u) s += hist[NBINS - 1 - (t * 32 + u)];
    csum[t] = s;
    __syncthreads();
    for (int off = 1; off < 512; off <<= 1) {
        unsigned v = csum[t];
        unsigned a = (t >= (unsigned)off) ? csum[t - off] : 0u;
        __syncthreads();
        csum[t] = v + a;
        __syncthreads();
    }
    unsigned incl = csum[t];
    unsigned excl = incl - s;
    if (excl < KTOP && incl >= KTOP) {
        unsigned run = excl, cb = 0;
        for (int u = 0; u < 32; ++u) {
            unsigned bin = NBINS - 1 - (t * 32 + u);
            run += hist[bin];
            if (run >= KTOP) { cb = bin; break; }
        }
        Wp[2] = cb;
    }
    if (t == 511 && incl < KTOP) Wp[2] = 0u; // fewer than K peaks: keep all
}

// ---------------- kernel 3: compact candidates with bin >= cutoff ----------
__global__ void __launch_bounds__(256) collect_kernel(unsigned* __restrict__ Wp,
                                                      unsigned cap) {
    unsigned n = Wp[0]; if (n > cap) n = cap;
    unsigned cb = Wp[2];
    const float* candV = (const float*)(Wp + WS_CAND);
    const unsigned* candI = Wp + WS_CAND + cap;
    float* selV = (float*)(Wp + WS_SELV);
    unsigned* selI = Wp + WS_SELI;
    for (unsigned i = blockIdx.x * blockDim.x + threadIdx.x; i < n;
         i += gridDim.x * blockDim.x) {
        float v = candV[i];
        int bin = (int)((v - 0.9f) * 163840.0f);
        bin = bin < 0 ? 0 : (bin > NBINS - 1 ? NBINS - 1 : bin);
        if ((unsigned)bin >= cb) {
            unsigned p = atomicAdd(Wp + 1, 1u);
            if (p < SELCAP) { selV[p] = v; selI[p] = candI[i]; }
        }
    }
}

// ---------------- kernel 4: bitonic sort 8192 keys in LDS, emit top-K ------
// key = (float_bits(value) << 32) | ~index  -> descending u64 sort gives
// value-descending, index-ascending tie-break (matches jax.lax.top_k).
__global__ void __launch_bounds__(1024) sort_emit_kernel(unsigned* __restrict__ Wp,
                                                         float* __restrict__ xyz,
                                                         float* __restrict__ valid) {
    const int N = SELCAP;
    __shared__ unsigned long long key[SELCAP]; // 64 KB
    const unsigned t = threadIdx.x;
    unsigned n = Wp[1]; if (n > SELCAP) n = SELCAP;
    const float* selV = (const float*)(Wp + WS_SELV);
    const unsigned* selI = Wp + WS_SELI;

    for (int i = t; i < N; i += 1024) {
        unsigned long long k = 0ULL;
        if (i < (int)n) {
            unsigned vb = __float_as_uint(selV[i]);
            k = ((unsigned long long)vb << 32) | (unsigned)(~selI[i]);
        }
        key[i] = k;
    }
    __syncthreads();

    for (int ksz = 2; ksz <= N; ksz <<= 1) {
        for (int j = ksz >> 1; j > 0; j >>= 1) {
            for (int i = t; i < N; i += 1024) {
                int l = i ^ j;
                if (l > i) {
                    unsigned long long a = key[i], b = key[l];
                    bool desc = ((i & ksz) == 0);
                    if (desc ? (a < b) : (a > b)) { key[i] = b; key[l] = a; }
                }
            }
            __syncthreads();
        }
    }

    for (int i = t; i < KTOP; i += 1024) {
        unsigned long long k = key[i];
        float vx = 0.0f, vy = 0.0f, vv = 0.0f;
        if (k != 0ULL) {
            unsigned idx = ~(unsigned)(k & 0xffffffffu);
            unsigned x = idx >> 12;      // / 4096
            unsigned y = idx & 4095u;    // % 4096
            vx = (float)x * 0.25f + 0.5f;
            vy = (float)y * 0.25f + 0.5f;
            vv = 1.0f;
        }
        xyz[2 * i] = vx;
        xyz[2 * i + 1] = vy;
        valid[i] = vv;
    }
}

// ---------------- host-side launcher ----------------
extern "C" void kernel_launch(void* const* d_in, const int* in_sizes, int n_in,
                              void* d_out, int out_size, void* d_ws, size_t ws_size,
                              hipStream_t stream) {
    const float* pred = (const float*)d_in[0];
    float* conf = (float*)d_out;                       // 4096*4096
    float* xyz = conf + (size_t)HH * WW;               // 4096*2
    float* valid = xyz + 2 * KTOP;                     // 4096
    unsigned* Wp = (unsigned*)d_ws;

    long long words = (long long)(ws_size / 4);
    long long capll = (words - WS_CAND) / 2;
    if (capll < 0) capll = 0;
    if (capll > (1 << 22)) capll = (1 << 22);
    unsigned cap = (unsigned)capll;

    init_kernel<<<dim3((WS_HIST + NBINS + 1023) / 1024), dim3(1024), 0, stream>>>(Wp);
    conf_kernel<<<dim3(WW / 128, HH / 32), dim3(256), 0, stream>>>(pred, conf, Wp, cap);
    cutoff_kernel<<<dim3(1), dim3(512), 0, stream>>>(Wp);
    collect_kernel<<<dim3(1024), dim3(256), 0, stream>>>(Wp, cap);
    sort_emit_kernel<<<dim3(1), dim3(1024), 0, stream>>>(Wp, xyz, valid);
}